// SSMLayer_15504831939340
// MI455X (gfx1250) — compile-verified
//
#include <hip/hip_runtime.h>

// Problem constants (from reference): N=64 state, H=256 channels, B=8, L=1024.
#define NS 64
#define HC 256
#define NB 8
#define LL 1024
#define MS 66   // padded LDS row stride for 64-wide matrices (bank spread)

typedef __attribute__((ext_vector_type(2))) float v2f;
typedef __attribute__((ext_vector_type(8))) float v8f;

// ---------------------------------------------------------------------------
// Workgroup-wide 64x64 @ 64x(16*NCT) matmul in LDS using v_wmma_f32_16x16x4.
// All matrices row-major with stride MS. 8 waves split the output tiles.
// Two-phase per tile: stage all 32 fragments (LDS loads pipeline), then fire
// 16 back-to-back WMMAs as a C-accumulation chain -- no per-WMMA dscnt stall.
// Fragment layouts (ISA 7.12.2, wave32):
//   A 16x4 f32 : VGPR0 = {K0 | K2} across lane halves, VGPR1 = {K1 | K3}
//   B 4x16 f32 : VGPR0 = {row0 | row2}, VGPR1 = {row1 | row3}  (mirror of A)
//   C/D 16x16  : VGPR r = rows (r | r+8) across lane halves
// ---------------------------------------------------------------------------
template <int NCT>
__device__ __forceinline__ void wg_matmul(float* __restrict__ D,
                                          const float* __restrict__ A,
                                          const float* __restrict__ Bm,
                                          int wave, int lr, int half)
{
  for (int tt = wave; tt < 4 * NCT; tt += 8) {       // wave-uniform trip
    const int rt = tt / NCT, ct = tt % NCT;
    const int r0 = rt * 16, c0 = ct * 16;

    v2f af[16], bf[16];
#pragma unroll
    for (int k = 0; k < 16; ++k) {                   // stage: 32 ds loads
      const int ka = 4 * k + 2 * half;
      af[k].x = A[(r0 + lr) * MS + ka];
      af[k].y = A[(r0 + lr) * MS + ka + 1];
      bf[k].x = Bm[(ka    ) * MS + c0 + lr];
      bf[k].y = Bm[(ka + 1) * MS + c0 + lr];
    }

    v8f acc = {0.f, 0.f, 0.f, 0.f, 0.f, 0.f, 0.f, 0.f};
#pragma unroll
    for (int k = 0; k < 16; ++k)                     // fire: 16 WMMAs
      acc = __builtin_amdgcn_wmma_f32_16x16x4_f32(false, af[k], false, bf[k],
                                                  (short)0, acc, false, false);
#pragma unroll
    for (int r = 0; r < 8; ++r)
      D[(r0 + r + 8 * half) * MS + c0 + lr] = acc[r];
  }
}

// ---------------------------------------------------------------------------
// Kernel 1 (fused): per-channel bilinear discretization (Gauss-Jordan, no
// pivoting -- I-sA is diagonally dominant) + K[h][l] = Cb Ab^l Bb via WMMA
// matrix-power doubling:
//   X[:,0..15] by 15 matvecs; R=Ab^16 (4 squarings); X[:,16..63] (3 products);
//   Q=Ab^64 (2 squarings); then 15x  X_{j+1} = Q X_j  (64^3 WMMA matmuls),
//   emitting 64 K values per block (overlapped with the matmul).
// Sequential depth: ~64 matvecs + 24 matmuls instead of a 1024-step scan.
// One workgroup (8 wave32) per channel. Dynamic LDS ~99KB (<320KB/WGP).
// ---------------------------------------------------------------------------
__global__ __launch_bounds__(256) void ssm_build_k(
    const float* __restrict__ As, const float* __restrict__ Bs,
    const float* __restrict__ Cs, const float* __restrict__ log_steps,
    float* __restrict__ Kws)
{
  extern __shared__ float smem[];
  float (*aug)[130] = (float (*)[130])smem;          // 64 x 130
  float* P  = smem + NS * 130;                       // 64 x MS
  float* Pn = P  + NS * MS;
  float* Xa = Pn + NS * MS;
  float* Xb = Xa + NS * MS;
  float* cb = Xb + NS * MS;                          // 64

  const int h    = blockIdx.x;
  const int t    = threadIdx.x;
  const int wave = t >> 5, lane = t & 31;
  const int half = lane >> 4, lr = lane & 15;

  // ---- phase A: Gauss-Jordan on [I-sA | I+sA | 2sB] (row x col-quarter) ----
  const int i  = t >> 2;                 // row owned
  const int q  = t & 3;                  // column quarter
  const int j0 = q * 33;
  const int j1 = (j0 + 33 < 130) ? j0 + 33 : 130;
  const float s = 0.5f * __expf(log_steps[h]);
  const float* Ah = As + (size_t)h * NS * NS;

  for (int j = j0; j < j1; ++j) {
    float v;
    if (j < NS)          { const float a = Ah[i * NS + j];
                           v = ((i == j) ? 1.f : 0.f) - s * a; }
    else if (j < 2 * NS) { const float a = Ah[i * NS + (j - NS)];
                           v = ((i == j - NS) ? 1.f : 0.f) + s * a; }
    else if (j == 2 * NS)  v = 2.f * s * Bs[h * NS + i];
    else                   v = 0.f;
    aug[i][j] = v;
  }
  if (t < NS) cb[t] = Cs[h * NS + t];
  __syncthreads();

  for (int p = 0; p < NS; ++p) {
    const float f = (i != p) ? aug[i][p] / aug[p][p] : 0.0f;
    __syncthreads();                       // all reads done before writes
    if (i != p)
      for (int j = j0; j < j1; ++j) aug[i][j] -= f * aug[p][j];
    __syncthreads();                       // writes done before next reads
  }

  // extract Ab -> P (normalize by surviving diagonal), Bb -> Xa column 0
  {
    const float dinv = 1.0f / aug[i][i];
    for (int j = j0; j < j1; ++j)
      if (j >= NS && j < 2 * NS) P[i * MS + (j - NS)] = aug[i][j] * dinv;
    if (q == 3) Xa[i * MS + 0] = aug[i][2 * NS] * dinv;
  }
  __syncthreads();

  // ---- phase B1: first 16 Krylov columns sequentially (cheap) ----
  for (int l = 1; l < 16; ++l) {
    if (t < NS) {
      float y = 0.f;
#pragma unroll
      for (int j = 0; j < NS; ++j) y += P[t * MS + j] * Xa[j * MS + (l - 1)];
      Xa[t * MS + l] = y;
    }
    __syncthreads();
  }

  // ---- phase B2: R = Ab^16 via 4 WMMA squarings (ping-pong P <-> Pn) ----
  float* Pa = P; float* Pb = Pn;
  for (int sq = 0; sq < 4; ++sq) {
    wg_matmul<4>(Pb, Pa, Pa, wave, lr, half);
    __syncthreads();
    float* tp = Pa; Pa = Pb; Pb = tp;
  }
  // ---- phase B3: expand X columns 16..63, three width-16 products ----
  for (int e = 0; e < 3; ++e) {
    wg_matmul<1>(Xa + 16 * (e + 1), Pa, Xa + 16 * e, wave, lr, half);
    __syncthreads();
  }
  // ---- phase B4: Q = Ab^64 via 2 more squarings ----
  for (int sq = 0; sq < 2; ++sq) {
    wg_matmul<4>(Pb, Pa, Pa, wave, lr, half);
    __syncthreads();
    float* tp = Pa; Pa = Pb; Pb = tp;
  }

  // ---- phase B5: 16 blocks of 64 K values; X_{j+1} = Q X_j via WMMA ----
  float* Xc = Xa; float* Xd = Xb;
  for (int blk = 0; blk < 16; ++blk) {
    if (t < NS) {                           // waves 0-1: emit K (reads Xc)
      float ks = 0.f;
#pragma unroll
      for (int r = 0; r < NS; ++r) ks += cb[r] * Xc[r * MS + t];
      Kws[(size_t)h * LL + blk * NS + t] = ks;
    }
    if (blk < 15)                           // all waves: matmul (writes Xd)
      wg_matmul<4>(Xd, Pa, Xc, wave, lr, half);
    __syncthreads();
    float* tx = Xc; Xc = Xd; Xd = tx;
  }
}

// ---------------------------------------------------------------------------
// Kernel 2: causal convolution as lower-triangular Toeplitz matmul:
//   Y_h (L x 8) = T_h (L x L) U_h,  T[l][m] = K[l-m].
// One workgroup (8 wave32) per channel. U padded to 16 zero columns and K
// padded with 16 leading zeros in LDS -> no boundary cndmasks. Each outer
// step stages 8 fragment pairs (16 taps) then fires 4 WMMAs back-to-back.
// Outer trip count is exactly t+1.
// ---------------------------------------------------------------------------
__global__ __launch_bounds__(256) void ssm_conv_wmma(
    const float* __restrict__ inp, const float* __restrict__ Kws,
    const float* __restrict__ Ds, float* __restrict__ out)
{
  extern __shared__ float smem[];
  float* Ush = smem;                 // [1024][16], cols 8..15 zero
  float* Kp  = smem + LL * 16;       // [16 + 1024], first 16 zero

  const int h    = blockIdx.x;
  const int tid  = threadIdx.x;
  const int wave = tid >> 5;
  const int lane = tid & 31;
  const int half = lane >> 4;
  const int lr   = lane & 15;

  for (int idx = tid; idx < LL * 16; idx += 256) {
    const int l = idx >> 4, c = idx & 15;
    Ush[idx] = (c < NB) ? inp[(size_t)c * LL * HC + (size_t)l * HC + h] : 0.f;
  }
  for (int idx = tid; idx < 16 + LL; idx += 256)
    Kp[idx] = (idx < 16) ? 0.f : Kws[(size_t)h * LL + (idx - 16)];
  __syncthreads();

  const float Dh = Ds[h];

  for (int t = wave; t < LL / 16; t += 8) {       // uniform per wave
    const int l0 = t * 16;
    v8f acc = {0.f, 0.f, 0.f, 0.f, 0.f, 0.f, 0.f, 0.f};

    for (int m0 = 0; m0 < l0 + 16; m0 += 16) {    // exactly t+1 trips
      v2f af[4], bf[4];
#pragma unroll
      for (int u = 0; u < 4; ++u) {               // stage 16 taps
        const int mrow = m0 + 4 * u + 2 * half;
        const int ia   = 16 + l0 + lr - mrow;     // always in-bounds
        af[u].x = Kp[ia];
        af[u].y = Kp[ia - 1];
        bf[u].x = Ush[mrow * 16 + lr];
        bf[u].y = Ush[(mrow + 1) * 16 + lr];
      }
#pragma unroll
      for (int u = 0; u < 4; ++u)                 // fire 4 WMMAs
        acc = __builtin_amdgcn_wmma_f32_16x16x4_f32(false, af[u], false, bf[u],
                                                    (short)0, acc, false, false);
    }

    if (lr < NB) {                                // cols 8..15 are padding
#pragma unroll
      for (int r = 0; r < 8; ++r) {
        const int row = l0 + r + 8 * half;        // D: VGPR r -> rows r, r+8
        const float uv = Ush[row * 16 + lr];
        out[(size_t)lr * LL * HC + (size_t)row * HC + h] = acc[r] + Dh * uv;
      }
    }
  }
}

// ---------------------------------------------------------------------------
extern "C" void kernel_launch(void* const* d_in, const int* in_sizes, int n_in,
                              void* d_out, int out_size, void* d_ws, size_t ws_size,
                              hipStream_t stream)
{
  const float* inp = (const float*)d_in[0];   // (B,L,H)
  const float* As  = (const float*)d_in[1];   // (H,N,N)
  const float* Bs  = (const float*)d_in[2];   // (H,N,1)
  const float* Cs  = (const float*)d_in[3];   // (H,1,N)
  const float* Ds  = (const float*)d_in[4];   // (H,)
  const float* ls  = (const float*)d_in[5];   // (H,)
  float* out = (float*)d_out;

  float* Kws = (float*)d_ws;                  // H*L floats (1 MB), channel-major

  const size_t lds1 = (size_t)(NS * 130 + 4 * NS * MS + NS) * sizeof(float); // ~99KB
  const size_t lds2 = (size_t)(LL * 16 + 16 + LL) * sizeof(float);           // ~68KB

  ssm_build_k  <<<dim3(HC), dim3(256), lds1, stream>>>(As, Bs, Cs, ls, Kws);
  ssm_conv_wmma<<<dim3(HC), dim3(256), lds2, stream>>>(inp, Kws, Ds, out);
}